// Render_46153718563035
// MI455X (gfx1250) — compile-verified
//
#include <hip/hip_runtime.h>
#include <cstdint>

typedef float v2f __attribute__((ext_vector_type(2)));
typedef float v8f __attribute__((ext_vector_type(8)));

#define SHARP 50.0f
#define UVS 49  // LDS row stride for u,v staging (gcd(49,64)=1 -> conflict-free)

__device__ __forceinline__ float sigm(float x) {
  return 1.0f / (1.0f + __expf(-x));
}

// One workgroup = one 16x16 block of the padded canvas (65x65 blocks).
// Each block is covered by at most 4 grid cells, one per parity class,
// composited in the reference's pass order: (e,e),(o,o),(o,e),(e,o).
__global__ __launch_bounds__(256)
void render_kernel(const float* __restrict__ param,
                   const int* __restrict__ decision,
                   const float* __restrict__ canvas,
                   float* __restrict__ out)
{
  __shared__ float sStk[20 * 12];       // raw params -> derived stroke consts
  __shared__ float sUV[256 * UVS];      // u,v per (pixel, stroke-component)

  const int by = blockIdx.y;            // padded row-block a in [0,65)
  const int bx = blockIdx.x;            // padded col-block in [0,65)
  const int t  = threadIdx.x;

  // Covering cells in composite (pass) order.
  const int ie = (by & 1) ? by - 1 : by;
  const int io = (by & 1) ? by : by - 1;
  const int je = (bx & 1) ? bx - 1 : bx;
  const int jo = (bx & 1) ? bx : bx - 1;
  const int ci[4] = { ie, io, io, ie };
  const int cj[4] = { je, jo, je, jo };
  int cells[4];
  int ncell = 0;
#pragma unroll
  for (int q = 0; q < 4; ++q) {
    if (ci[q] >= 0 && ci[q] < 64 && cj[q] >= 0 && cj[q] < 64)
      cells[ncell++] = ci[q] * 64 + cj[q];
  }
  const int nstr = ncell * 5;           // <= 20 strokes, uniform per block

  // ---- Phase 0: async-copy raw stroke params (12 floats each) into LDS ----
  if (t < nstr * 3) {
    const int q  = t / 3;
    const int f  = (t - q * 3) * 4;     // float offset 0/4/8 (16B chunks)
    const int cq = q / 5;
    const int k  = q - cq * 5;
    const int gs = cells[cq] * 5 + k;   // global stroke index
    const float* g = param + (size_t)gs * 12 + f;
    const uint32_t lds = (uint32_t)(size_t)(&sStk[q * 12 + f]);
    asm volatile("global_load_async_to_lds_b128 %0, %1, off"
                 :: "v"(lds), "v"(g) : "memory");
  }
  asm volatile("s_wait_asynccnt 0" ::: "memory");
  __syncthreads();

  // ---- Phase 1: derive per-stroke constants in place ----
  if (t < nstr) {
    const int cq = t / 5;
    const int k  = t - cq * 5;
    const int gcell = cells[cq];
    const int i = gcell >> 6, j = gcell & 63;
    const int gs = gcell * 5 + k;
    const float p0 = sigm(sStk[t * 12 + 0]);
    const float p1 = sigm(sStk[t * 12 + 1]);
    const float p2 = sigm(sStk[t * 12 + 2]);
    const float p3 = sigm(sStk[t * 12 + 3]);
    const float p4 = sigm(sStk[t * 12 + 4]);
    const float cr = sigm(sStk[t * 12 + 5]);
    const float cg = sigm(sStk[t * 12 + 6]);
    const float cb = sigm(sStk[t * 12 + 7]);
    // Fold the cell origin into the stroke center: gx = X - j/2, gy = Y - i/2.
    const float x0 = p0 + 0.5f * (float)j;
    const float y0 = p1 + 0.5f * (float)i;
    const float hw = 0.5f * fmaxf(p2, 0.01f);
    const float hh = 0.5f * fmaxf(p3, 0.01f);
    const float th = p4 * 3.14159265358979323846f;
    float st, ct;
    __sincosf(th, &st, &ct);
    const float cu = -(x0 * ct + y0 * st);   // u = X*ct + Y*st + cu
    const float cv =  x0 * st - y0 * ct;     // v = -X*st + Y*ct + cv
    const float de = (float)decision[gs];
    sStk[t * 12 + 0] = ct; sStk[t * 12 + 1] = st;
    sStk[t * 12 + 2] = cu; sStk[t * 12 + 3] = cv;
    sStk[t * 12 + 4] = hw; sStk[t * 12 + 5] = hh;
    sStk[t * 12 + 6] = de;
    sStk[t * 12 + 7] = cr; sStk[t * 12 + 8] = cg; sStk[t * 12 + 9] = cb;
  }
  __syncthreads();

  // ---- Phase 2: u,v for all 256 pixels x 40 stroke-components via WMMA ----
  // [256 x 3] @ [3 x 48] with V_WMMA_F32_16X16X4_F32 (K padded to 4).
  // A(16x4): lanes 0-15 hold M=lane, VGPR0=K0(X), VGPR1=K1(Y);
  //          lanes 16-31 hold VGPR0=K2(=1), VGPR1=K3(=0).
  // B(4x16): lanes 0-15 rows K0,K1; lanes 16-31 rows K2,K3 (mirror of A).
  const int  L  = t & 31;
  const int  wv = t >> 5;               // 8 waves; each does 2 M-tiles x 3 N-tiles
  const int  n  = L & 15;
  const bool hi = (L >= 16);
  const float inv32 = 1.0f / 32.0f;

#pragma unroll
  for (int nt = 0; nt < 3; ++nt) {
    const int col = nt * 16 + n;        // stroke-component column (2s=u, 2s+1=v)
    float b0 = 0.0f, b1 = 0.0f, b2 = 0.0f;
    if (col < 2 * nstr) {
      const int s = col >> 1;
      const float ct = sStk[s * 12 + 0];
      const float st = sStk[s * 12 + 1];
      if (col & 1) { b0 = -st; b1 = ct; b2 = sStk[s * 12 + 3]; }
      else         { b0 =  ct; b1 = st; b2 = sStk[s * 12 + 2]; }
    }
    v2f bv;
    bv.x = hi ? b2 : b0;
    bv.y = hi ? 0.0f : b1;
#pragma unroll
    for (int q = 0; q < 2; ++q) {
      const int mt = wv * 2 + q;        // M-tile == pixel row within block
      v2f av;
      av.x = hi ? 1.0f : ((float)(16 * bx + n)  + 0.5f) * inv32;  // X
      av.y = hi ? 0.0f : ((float)(16 * by + mt) + 0.5f) * inv32;  // Y
      v8f d = {};
      d = __builtin_amdgcn_wmma_f32_16x16x4_f32(false, av, false, bv,
                                                (short)0, d, false, false);
#pragma unroll
      for (int r = 0; r < 8; ++r) {
        const int M = r + (hi ? 8 : 0); // D: VGPR r holds M=r / M=r+8
        sUV[(mt * 16 + M) * UVS + col] = d[r];
      }
    }
  }
  __syncthreads();

  // ---- Phase 3: per-pixel sequential over-composite of up to 20 strokes ----
  const int rl = t >> 4, cl = t & 15;
  const int y = 16 * by + rl - 8;       // unpad
  const int x = 16 * bx + cl - 8;
  if (y >= 0 && y < 1024 && x >= 0 && x < 1024) {
    const size_t base = (size_t)y * 1024 + x;
    float c0 = canvas[base];
    float c1 = canvas[base + 1048576];
    float c2 = canvas[base + 2097152];
    const float* uvp = &sUV[t * UVS];
    for (int s = 0; s < nstr; ++s) {
      const float u  = uvp[2 * s];
      const float v  = uvp[2 * s + 1];
      const float hw = sStk[s * 12 + 4];
      const float hh = sStk[s * 12 + 5];
      const float al = sigm(SHARP * (hw - fabsf(u))) *
                       sigm(SHARP * (hh - fabsf(v))) * sStk[s * 12 + 6];
      // c = color*al + c*(1-al) == c + al*(color - c)
      c0 = fmaf(al, sStk[s * 12 + 7] - c0, c0);
      c1 = fmaf(al, sStk[s * 12 + 8] - c1, c1);
      c2 = fmaf(al, sStk[s * 12 + 9] - c2, c2);
    }
    out[base]           = c0;
    out[base + 1048576] = c1;
    out[base + 2097152] = c2;
  }
}

extern "C" void kernel_launch(void* const* d_in, const int* in_sizes, int n_in,
                              void* d_out, int out_size, void* d_ws, size_t ws_size,
                              hipStream_t stream) {
  (void)in_sizes; (void)n_in; (void)out_size; (void)d_ws; (void)ws_size;
  const float* param    = (const float*)d_in[0];
  const int*   decision = (const int*)d_in[1];
  const float* canvas   = (const float*)d_in[2];
  float*       out      = (float*)d_out;
  dim3 grid(65, 65);
  render_kernel<<<grid, 256, 0, stream>>>(param, decision, canvas, out);
}